// IM_Layer_54168127537564
// MI455X (gfx1250) — compile-verified
//
#include <hip/hip_runtime.h>

typedef __attribute__((ext_vector_type(2)))  float    v2f;
typedef __attribute__((ext_vector_type(8)))  float    v8f;

#define EPS_DEN  1e-10f
#define EPS_FRAC 1e-7f

__device__ __forceinline__ float flog2(float x) { return __builtin_amdgcn_logf(x); }  // v_log_f32 (base-2)
__device__ __forceinline__ float frcp(float x)  { return __builtin_amdgcn_rcpf(x); }  // v_rcp_f32

// One thread = one neuron. p layout: (16,16,2,N), N innermost -> coalesced b32 loads.
// Register budget is the key constraint: only m12[16][2] (32 VGPRs) lives across the
// kernel; everything x-indexed is reduced to scalars inside a ROLLED x-loop so the
// compiler cannot hoist 512 loads into one live range (round-1 version spilled 940B).
__global__ __launch_bounds__(256) void pid_wmma_kernel(
    const float* __restrict__ p, float* __restrict__ out, int N)
{
  const int n = blockIdx.x * blockDim.x + threadIdx.x;
  if (n >= N) return;                 // N % 32 == 0 -> wave-uniform exit, EXEC stays all-1s for WMMA
  const size_t sN = (size_t)N;
  const float* __restrict__ base = p + n;

  // ---------- Pass A: m12[y][z] only (m1, m2 derive from it) ----------
  float m12_0[16], m12_1[16];
  #pragma unroll
  for (int i = 0; i < 16; ++i) { m12_0[i] = 0.f; m12_1[i] = 0.f; }

  #pragma unroll 1
  for (int x = 0; x < 16; ++x) {
    const float* qx = base + (size_t)(x * 32) * sN;
    #pragma unroll
    for (int y = 0; y < 16; ++y) {
      m12_0[y] += qx[(size_t)(2 * y) * sN];
      m12_1[y] += qx[(size_t)(2 * y + 1) * sN];
    }
  }
  float m2z0 = 0.f, m2z1 = 0.f;
  #pragma unroll
  for (int i = 0; i < 16; ++i) { m2z0 += m12_0[i]; m2z1 += m12_1[i]; }

  // ---------- Pass B: per-x row resident in registers; red/syn/unq0 terms ----------
  float accR = 0.f, accS = 0.f, acc1 = 0.f;
  #pragma unroll 1
  for (int x = 0; x < 16; ++x) {
    const float* qx = base + (size_t)(x * 32) * sN;
    float pv0[16], pv1[16];
    #pragma unroll
    for (int y = 0; y < 16; ++y) {
      pv0[y] = qx[(size_t)(2 * y) * sN];
      pv1[y] = qx[(size_t)(2 * y + 1) * sN];
    }
    // m02[x][z] as scalars
    float m02x0 = 0.f, m02x1 = 0.f;
    #pragma unroll
    for (int y = 0; y < 16; ++y) { m02x0 += pv0[y]; m02x1 += pv1[y]; }
    const float m0x = m02x0 + m02x1;

    // I_{(0,)} contribution of this x (frac independent of y; weight = m02[x][z])
    {
      float f1 = m02x0 * frcp(m0x * m2z0 + EPS_DEN);
      f1 = (f1 < EPS_FRAC) ? 1.0f : f1;
      acc1 += m02x0 * flog2(f1);
      float f2 = m02x1 * frcp(m0x * m2z1 + EPS_DEN);
      f2 = (f2 < EPS_FRAC) ? 1.0f : f2;
      acc1 += m02x1 * flog2(f2);
    }

    #pragma unroll
    for (int y = 0; y < 16; ++y) {
      const float p0  = pv0[y];
      const float p1  = pv1[y];
      const float m1y = m12_0[y] + m12_1[y];
      const float m01 = p0 + p1;              // m[(0,1)][x][y] on the fly
      const float dR  = m0x + m1y - m01;      // union prob for {(0),(1)}
      // ---- z = 0 ----
      {
        float fS = p0 * frcp(m01 * m2z0 + EPS_DEN);
        fS = (fS < EPS_FRAC) ? 1.0f : fS;
        accS += p0 * flog2(fS);
        const float num = m02x0 + m12_0[y] - p0;
        float fR = num * frcp(dR * m2z0 + EPS_DEN);
        fR = (fR < EPS_FRAC) ? 1.0f : fR;
        accR += p0 * flog2(fR);
      }
      // ---- z = 1 ----
      {
        float fS = p1 * frcp(m01 * m2z1 + EPS_DEN);
        fS = (fS < EPS_FRAC) ? 1.0f : fS;
        accS += p1 * flog2(fS);
        const float num = m02x1 + m12_1[y] - p1;
        float fR = num * frcp(dR * m2z1 + EPS_DEN);
        fR = (fR < EPS_FRAC) ? 1.0f : fR;
        accR += p1 * flog2(fR);
      }
    }
  }

  // ---------- I_{(1,)}: collapses over x; 32 terms from m12 registers ----------
  float acc2 = 0.f;
  #pragma unroll
  for (int i = 0; i < 16; ++i) {
    const float m1i = m12_0[i] + m12_1[i];
    float f1 = m12_0[i] * frcp(m1i * m2z0 + EPS_DEN);
    f1 = (f1 < EPS_FRAC) ? 1.0f : f1;
    acc2 += m12_0[i] * flog2(f1);
    float f2 = m12_1[i] * frcp(m1i * m2z1 + EPS_DEN);
    f2 = (f2 < EPS_FRAC) ? 1.0f : f2;
    acc2 += m12_1[i] * flog2(f2);
  }

  const float Htot = -(m2z0 * flog2(m2z0 + EPS_DEN) + m2z1 * flog2(m2z1 + EPS_DEN));
  const float I0 = accR, I1 = acc1, I2 = acc2, I3 = accS;

  // ---------- Moebius transform Pi = M(4x4) @ I_cap via V_WMMA_F32_16X16X4_F32 ----------
  // A (16x4 f32, 2 VGPR): lanes 0-15 = rows, vgpr{0,1} = K{0,1}; lanes 16-31: K{2,3}.
  // B (4x16 f32, 2 VGPR): lanes 0-15 = cols(N), vgpr{0,1} = K{0,1}; lanes 16-31: K{2,3}.
  // D (16x16 f32, 8 VGPR): vgpr r, lanes 0-15 -> row r.  Rows 0-3 carry Pi.
  const unsigned lane = threadIdx.x & 31u;
  const unsigned row  = lane & 15u;
  const bool     hi   = lane >= 16u;

  // Moebius columns (A rows 0..3; rows 4..15 zero)
  const float e0 = (row==0u || row==3u) ? 1.f : ((row==1u || row==2u) ? -1.f : 0.f); // x I0
  const float e1 = (row==1u) ? 1.f : ((row==3u) ? -1.f : 0.f);                       // x I1
  const float e2 = (row==2u) ? 1.f : ((row==3u) ? -1.f : 0.f);                       // x I2
  const float e3 = (row==3u) ? 1.f : 0.f;                                            // x I3

  v8f D0, D1;
  const int src0 = (int)row, src1 = 16 + (int)row;
  {
    v2f A;  A[0] = hi ? e2 : e0;  A[1] = hi ? e3 : e1;
    { // neurons on lanes 0..15
      float t0=__shfl(I0,src0,32), t1=__shfl(I1,src0,32), t2=__shfl(I2,src0,32), t3=__shfl(I3,src0,32);
      v2f B;  B[0] = hi ? t2 : t0;  B[1] = hi ? t3 : t1;
      v8f C = {};
      D0 = __builtin_amdgcn_wmma_f32_16x16x4_f32(false, A, false, B, (short)0, C, false, false);
    }
    { // neurons on lanes 16..31
      float t0=__shfl(I0,src1,32), t1=__shfl(I1,src1,32), t2=__shfl(I2,src1,32), t3=__shfl(I3,src1,32);
      v2f B;  B[0] = hi ? t2 : t0;  B[1] = hi ? t3 : t1;
      v8f C = {};
      D1 = __builtin_amdgcn_wmma_f32_16x16x4_f32(false, A, false, B, (short)0, C, false, false);
    }
  }

  // Regather lane-private Pi and store coalesced rows of the (5,N) output.
  const size_t nn = (size_t)n;
  #pragma unroll
  for (int r = 0; r < 4; ++r) {
    const float fromHi = __shfl(D1[r], (int)row, 32);
    const float pi = hi ? fromHi : D0[r];
    out[(size_t)r * sN + nn] = pi;
  }
  out[(size_t)4 * sN + nn] = Htot - I3;   // H_res
}

extern "C" void kernel_launch(void* const* d_in, const int* in_sizes, int n_in,
                              void* d_out, int out_size, void* d_ws, size_t ws_size,
                              hipStream_t stream) {
  const float* p = (const float*)d_in[0];
  float* out = (float*)d_out;
  const int N = in_sizes[0] / (16 * 16 * 2);   // 100000
  const int threads = 256;                      // 8 wave32s
  const int blocks = (N + threads - 1) / threads;
  pid_wmma_kernel<<<blocks, threads, 0, stream>>>(p, out, N);
}